// Expansion_33560874451632
// MI455X (gfx1250) — compile-verified
//
#include <hip/hip_runtime.h>

typedef float v2f __attribute__((ext_vector_type(2)));
typedef float v8f __attribute__((ext_vector_type(8)));
typedef int   v4i __attribute__((vector_size(16)));

#define AS1 __attribute__((address_space(1)))
#define AS3 __attribute__((address_space(3)))

#define DIM_IN   576
#define NV4      144      // DIM_IN / 4
#define NW       1929     // NUM_WEIGHTS
#define NWP      1936     // padded to 121 tiles of 16
#define NTILES   121
#define BT       16       // batch rows per block
#define XS       580      // padded x row stride in LDS (580*4 B = 16B aligned)
#define RS       96       // padded res row stride
#define NRES     90
#define NOUTEL   169
#define BLOCK    128

#if defined(__has_builtin)
#if __has_builtin(__builtin_amdgcn_global_load_async_to_lds_b128)
#define USE_ASYNC_LDS 1
#endif
#endif

// ------------------------- compile-time tables -------------------------

struct InstMeta { int i, j, k, U, V, wo, ebase, biasbase; };

constexpr InstMeta INSTS[11] = {
  {0,0,0, 2,2,    0,   0, 1920},
  {0,1,1, 2,2,  256,   4, 1924},
  {0,2,2, 1,1,  512,   8, 1928},
  {1,0,1, 2,2,  576,   9,   -1},
  {1,1,0, 2,2,  832,  21,   -1},
  {1,1,2, 2,1, 1088,  33,   -1},
  {1,2,1, 1,2, 1216,  39,   -1},
  {2,0,2, 2,1, 1344,  45,   -1},
  {2,1,1, 2,2, 1472,  55,   -1},
  {2,2,0, 1,2, 1728,  75,   -1},
  {2,2,2, 1,1, 1856,  85,   -1},
};

// Wigner 3j (1,1,2), real basis l=1:{y,z,x}, l=2:{xy,yz,z2,xz,x2-y2},
// Frobenius-normalized (Gaunt-derived). Symmetric in (a,b).
constexpr double G112(int a, int b, int c) {
  const double S  = 0.31622776601683794;   // 1/sqrt(10)
  const double A  = 0.18257418583505536;   // 1/sqrt(30)
  const double B2 = 0.36514837167011072;   // 2/sqrt(30)
  if (c == 0) return ((a == 2 && b == 0) || (a == 0 && b == 2)) ? S : 0.0;
  if (c == 1) return ((a == 0 && b == 1) || (a == 1 && b == 0)) ? S : 0.0;
  if (c == 3) return ((a == 2 && b == 1) || (a == 1 && b == 2)) ? S : 0.0;
  if (c == 2) { if (a == b) { if (a == 1) return B2; return -A; } return 0.0; }
  if (c == 4) { if (a == b) { if (a == 2) return S; if (a == 0) return -S; } return 0.0; }
  return 0.0;
}
// Wigner 3j (2,2,2), fully symmetric, Frobenius-normalized.
constexpr double G222(int a, int b, int c) {
  const double U = 0.20701966780270626;   // sqrt(3/70)
  const double P = 0.11952286093343936;   // 1/sqrt(70)
  const double Q = 0.23904572186687872;   // 2/sqrt(70)
  int x = a, y = b, z = c, t = 0;
  if (x > y) { t = x; x = y; y = t; }
  if (y > z) { t = y; y = z; z = t; }
  if (x > y) { t = x; x = y; y = t; }
  if (x == 0 && y == 1 && z == 3) return U;
  if (x == 0 && y == 0 && z == 2) return -Q;
  if (x == 1 && y == 1 && z == 2) return P;
  if (x == 2 && y == 3 && z == 3) return P;
  if (x == 1 && y == 1 && z == 4) return -U;
  if (x == 3 && y == 3 && z == 4) return U;
  if (x == 2 && y == 2 && z == 2) return Q;
  if (x == 2 && y == 4 && z == 4) return -Q;
  return 0.0;
}
constexpr double w3jval(int m, int i1, int i2, int k3) {
  const double D3 = 0.57735026918962584;  // 1/sqrt(3)
  const double D5 = 0.44721359549995793;  // 1/sqrt(5)
  switch (m) {
    case 0:  return 1.0;
    case 1:  return (i1 == i2) ? D3 : 0.0;
    case 2:  return (i1 == i2) ? D5 : 0.0;
    case 3:  return (i2 == k3) ? D3 : 0.0;
    case 4:  return (i1 == k3) ? D3 : 0.0;
    case 5:  return G112(i1, k3, i2);    // wigner(1,2,1)
    case 6:  return G112(i2, k3, i1);    // wigner(2,1,1)
    case 7:  return (i2 == k3) ? D5 : 0.0;
    case 8:  return G112(i1, i2, k3);    // wigner(1,1,2)
    case 9:  return (i1 == k3) ? D5 : 0.0;
    case 10: return G222(i1, i2, k3);
  }
  return 0.0;
}

struct ResDesc {
  unsigned short col0;    // wb column for w=0
  unsigned short xbase;   // x row offset (xoff_i + k3)
  short bias;             // wb bias column or -1
  unsigned char wstride;  // = U*V
  unsigned char xstride;  // = dim(l_i)
};
struct ResTable { ResDesc d[NRES]; };

constexpr ResTable make_res() {
  ResTable t{};
  constexpr int XOFF[3] = {0, 64, 256};
  constexpr int XDIM[3] = {1, 3, 5};
  for (int m = 0; m < 11; ++m) {
    InstMeta I = INSTS[m];
    int D = XDIM[I.i], UV = I.U * I.V;
    for (int u = 0; u < I.U; ++u)
      for (int vv = 0; vv < I.V; ++vv)
        for (int k3 = 0; k3 < D; ++k3) {
          int e = I.ebase + (u * I.V + vv) * D + k3;
          t.d[e].col0    = (unsigned short)(I.wo + u * I.V + vv);
          t.d[e].wstride = (unsigned char)UV;
          t.d[e].xbase   = (unsigned short)(XOFF[I.i] + k3);
          t.d[e].xstride = (unsigned char)D;
          t.d[e].bias    = (short)(I.biasbase >= 0 ? I.biasbase + u * I.V + vv : -1);
        }
  }
  return t;
}

struct OutTable { short offs[NOUTEL + 1]; short residx[256]; float coeff[256]; };

constexpr OutTable make_out() {
  OutTable t{};
  constexpr int XDIM[3] = {1, 3, 5};
  constexpr int ODIM[3] = {1, 3, 5};
  constexpr int ROFF[3] = {0, 2, 8};
  short outi[256] = {}; short ri[256] = {}; double cf[256] = {};
  int n = 0;
  for (int m = 0; m < 11; ++m) {
    InstMeta I = INSTS[m];
    int d1 = ODIM[I.j], d2 = ODIM[I.k], Din = XDIM[I.i];
    for (int i1 = 0; i1 < d1; ++i1)
      for (int i2 = 0; i2 < d2; ++i2)
        for (int k3 = 0; k3 < Din; ++k3) {
          double val = w3jval(m, i1, i2, k3);
          if (val == 0.0) continue;
          for (int u = 0; u < I.U; ++u)
            for (int vv = 0; vv < I.V; ++vv) {
              int row = ROFF[I.j] + u * d1 + i1;
              int col = ROFF[I.k] + vv * d2 + i2;
              outi[n] = (short)(row * 13 + col);
              ri[n]   = (short)(I.ebase + (u * I.V + vv) * Din + k3);
              // folds w3j * (1/mw) * (1/sqrt(MUL0)) = val / (64 * 8)
              cf[n]   = val / 512.0;
              ++n;
            }
        }
  }
  int cnt[NOUTEL] = {};
  for (int e = 0; e < n; ++e) cnt[outi[e]]++;
  t.offs[0] = 0;
  for (int o = 0; o < NOUTEL; ++o) t.offs[o + 1] = (short)(t.offs[o] + cnt[o]);
  int pos[NOUTEL] = {};
  for (int o = 0; o < NOUTEL; ++o) pos[o] = t.offs[o];
  for (int e = 0; e < n; ++e) {
    int o = outi[e]; int p = pos[o]++;
    t.residx[p] = ri[e];
    t.coeff[p]  = (float)cf[e];
  }
  return t;
}

__constant__ ResTable RES_TAB = make_res();
__constant__ OutTable OUT_TAB = make_out();

// ------------------------------- kernel --------------------------------

__global__ __launch_bounds__(BLOCK)
void expansion_kernel(const float* __restrict__ x, const float* __restrict__ W,
                      float* __restrict__ out, int B) {
  extern __shared__ float smem[];
  float* xs  = smem;              // BT * XS
  float* wb  = xs + BT * XS;      // BT * NWP   (unscaled x0@W; scale folded in coeffs)
  float* res = wb + BT * NWP;     // BT * RS

  const int tid = threadIdx.x;
  const int b0  = blockIdx.x * BT;

  // ---- phase 0: stage 16 contiguous input rows into LDS (b128 granularity) ----
#if USE_ASYNC_LDS
  if (b0 + BT <= B) {
    // async DMA: global -> LDS, no VGPR round trip, tracked by ASYNCcnt
    const float* src = x + (size_t)b0 * DIM_IN;   // 16 rows are contiguous
    for (int e = tid; e < BT * NV4; e += BLOCK) {
      int r = e / NV4, f = e - r * NV4;
      __builtin_amdgcn_global_load_async_to_lds_b128(
          (AS1 v4i*)(src + (size_t)e * 4),
          (AS3 v4i*)(xs + r * XS + f * 4),
          0, 0);
    }
    asm volatile("s_wait_asynccnt 0x0" ::: "memory");
  } else
#endif
  {
    for (int e = tid; e < BT * NV4; e += BLOCK) {
      int r = e / NV4, f = e - r * NV4;
      int b = b0 + r;
      float4 vv = make_float4(0.f, 0.f, 0.f, 0.f);
      if (b < B) vv = ((const float4*)x)[(size_t)b * NV4 + f];
      *(float4*)(xs + r * XS + f * 4) = vv;
    }
  }
  __syncthreads();

  // ---- phase 1: wb[16 x 1929] = x0 @ W_lin via V_WMMA_F32_16X16X4_F32 ----
  const int wave  = tid >> 5;
  const int lane  = tid & 31;
  const int laneM = lane & 15;
  const int laneH = lane >> 4;

  // A-frag (16x4 f32): lanes 0-15 hold K={0,1}, lanes 16-31 hold K={2,3}
  v2f a[16];
#pragma unroll
  for (int kc = 0; kc < 16; ++kc) {
    int kk = 4 * kc + 2 * laneH;
    a[kc].x = xs[laneM * XS + kk];
    a[kc].y = xs[laneM * XS + kk + 1];
  }

  for (int nt = wave; nt < NTILES; nt += 4) {
    const int n0 = nt * 16;
    // clamp instead of predicate: wb padding columns are never consumed,
    // so duplicated column NW-1 data is harmless and EXEC stays uniform.
    int col = n0 + laneM;
    col = (col < NW) ? col : (NW - 1);
    const int pnt = (nt + 4 < NTILES) ? (nt + 4) : nt;   // wave-uniform
    __builtin_prefetch(&W[(2 * lane) * NW + pnt * 16], 0, 3);
    v8f acc = {0.f, 0.f, 0.f, 0.f, 0.f, 0.f, 0.f, 0.f};
#pragma unroll
    for (int kc = 0; kc < 16; ++kc) {
      const int kr = 4 * kc + 2 * laneH;
      v2f bf;
      bf.x = W[kr * NW + col];
      bf.y = W[(kr + 1) * NW + col];
      acc = __builtin_amdgcn_wmma_f32_16x16x4_f32(false, a[kc], false, bf,
                                                  (short)0, acc, false, false);
    }
    // C-frag (16x16 f32): VGPR g -> row M = g + 8*laneH, col N = laneM
#pragma unroll
    for (int g = 0; g < 8; ++g)
      wb[(g + 8 * laneH) * NWP + n0 + laneM] = acc[g];
  }
  __syncthreads();

  // ---- phase 2: res[b,e] = bias + sum_w wb[b, col0 + w*UV] * x[b, xbase + w*D] ----
  for (int q = tid; q < BT * NRES; q += BLOCK) {
    int r = q / NRES, e = q - r * NRES;
    ResDesc d = RES_TAB.d[e];
    const float* wr = &wb[r * NWP + d.col0];
    const float* xr = &xs[r * XS + d.xbase];
    float acc = (d.bias >= 0) ? wb[r * NWP + d.bias] : 0.0f;
    const int ws = d.wstride, xst = d.xstride;
#pragma unroll 4
    for (int w = 0; w < 64; ++w)
      acc += wr[w * ws] * xr[w * xst];
    res[r * RS + e] = acc;
  }
  __syncthreads();

  // ---- phase 3: out[b,o] = sum over CSR of (w3j/512) * res ----
  for (int q = tid; q < BT * NOUTEL; q += BLOCK) {
    int r = q / NOUTEL, o = q - r * NOUTEL;
    int b = b0 + r;
    if (b >= B) continue;
    float acc = 0.0f;
    for (int e = OUT_TAB.offs[o]; e < OUT_TAB.offs[o + 1]; ++e)
      acc += OUT_TAB.coeff[e] * res[r * RS + OUT_TAB.residx[e]];
    out[(long long)b * NOUTEL + o] = acc;
  }
}

// ------------------------------ launcher -------------------------------

extern "C" void kernel_launch(void* const* d_in, const int* in_sizes, int n_in,
                              void* d_out, int out_size, void* d_ws, size_t ws_size,
                              hipStream_t stream) {
  const float* x = (const float*)d_in[0];
  const float* W = (const float*)d_in[1];
  float* out = (float*)d_out;
  const int B = in_sizes[0] / DIM_IN;
  const int nblk = (B + BT - 1) / BT;
  const size_t smem = (size_t)(BT * XS + BT * NWP + BT * RS) * sizeof(float); // ~167 KB

  (void)hipFuncSetAttribute((const void*)expansion_kernel,
                            hipFuncAttributeMaxDynamicSharedMemorySize, (int)smem);
  expansion_kernel<<<nblk, BLOCK, smem, stream>>>(x, W, out, B);
}